// PostProcess_7791070675703
// MI455X (gfx1250) — compile-verified
//
#include <hip/hip_runtime.h>
#include <hip/hip_bf16.h>
#include <stdint.h>

// Problem constants (from the reference): B = 16384, N = 512, NUM = 3.
#define N_Q            512
#define NUM_VP         3
#define WAVES_PER_BLK  8
#define ITEMS_PER_LANE (N_Q / 32)          // 16
#define BYTES_PER_IMG  (N_Q * 3 * 4)       // 6144 B of positions per image
#define ASYNC_CHUNKS   (BYTES_PER_IMG / (32 * 16))  // 12 x b128 per lane

// cos(pi / (NUM+1)) = cos(pi/4): angle >= pi/4  <=>  |dot| <= CTH
#define CTH 0.70710678118654752f

__device__ __forceinline__ void wave_argmax(float& v, int& i) {
    // Full butterfly so every lane ends with the identical (max, lowest-index).
    #pragma unroll
    for (int m = 16; m >= 1; m >>= 1) {
        float v2 = __shfl_xor(v, m, 32);
        int   i2 = __shfl_xor(i, m, 32);
        if (v2 > v || (v2 == v && i2 < i)) { v = v2; i = i2; }
    }
}

__global__ __launch_bounds__(WAVES_PER_BLK * 32)
void vp_greedy_select_kernel(const float* __restrict__ logits,   // [B, N, 1]
                             const float* __restrict__ pos,      // [B, N, 3]
                             float* __restrict__ out,            // [B, 3, 3]
                             int B) {
    __shared__ __align__(16) float lpos[WAVES_PER_BLK * N_Q * 3];

    const int lane = threadIdx.x & 31;
    const int wave = threadIdx.x >> 5;
    const int b    = blockIdx.x * WAVES_PER_BLK + wave;
    if (b >= B) return;   // uniform per wave; no cross-wave sync anywhere

    float*       wpos = &lpos[wave * N_Q * 3];
    const float* gpos = pos    + (size_t)b * N_Q * 3;
    const float* glog = logits + (size_t)b * N_Q;

    // ---- CDNA5 async copy: stage this image's 6144B position block to LDS.
    // 12 x b128 per lane; tracked with ASYNCcnt (no VGPR staging).
    {
        const uint32_t lbase = (uint32_t)(uintptr_t)wpos;   // generic->LDS: low 32 bits
        const uint64_t gbase = (uint64_t)(uintptr_t)gpos;
        #pragma unroll
        for (int j = 0; j < ASYNC_CHUNKS; ++j) {
            const uint32_t off   = (uint32_t)(lane + 32 * j) * 16u;
            const uint32_t laddr = lbase + off;
            const uint64_t gaddr = gbase + off;
            asm volatile("global_load_async_to_lds_b128 %0, %1, off"
                         :: "v"(laddr), "v"(gaddr)
                         : "memory");
        }
    }

    // ---- Confidences (coalesced; overlaps with the async DMA in flight).
    // Lane owns items lane, lane+32, ... (stride-32 interleave).
    float c[ITEMS_PER_LANE];
    #pragma unroll
    for (int k = 0; k < ITEMS_PER_LANE; ++k)
        c[k] = glog[lane + 32 * k];

    asm volatile("s_wait_asynccnt 0" ::: "memory");

    // ---- Snapshot this lane's positions to registers.
    // Per-lane LDS address stride = 3 dwords; gcd(3,64)=1 => bank-conflict-free.
    float X[ITEMS_PER_LANE], Y[ITEMS_PER_LANE], Z[ITEMS_PER_LANE];
    #pragma unroll
    for (int k = 0; k < ITEMS_PER_LANE; ++k) {
        const int base = (lane + 32 * k) * 3;
        X[k] = wpos[base + 0];
        Y[k] = wpos[base + 1];
        Z[k] = wpos[base + 2];
    }

    const float NEG_INF = -__builtin_inff();
    int cand0, cand1 = 0, cand2 = 0;

    // ---- Pass 0: global argmax of confidence.
    float bv = NEG_INF; int bi = 0;
    #pragma unroll
    for (int k = 0; k < ITEMS_PER_LANE; ++k)
        if (c[k] > bv) { bv = c[k]; bi = lane + 32 * k; }
    wave_argmax(bv, bi);
    cand0 = bi;

    // Filter vs cand0 (angle < pi/4 => dead) and kill the selected item.
    {
        const float px = wpos[cand0 * 3 + 0];
        const float py = wpos[cand0 * 3 + 1];
        const float pz = wpos[cand0 * 3 + 2];
        #pragma unroll
        for (int k = 0; k < ITEMS_PER_LANE; ++k) {
            const int   i = lane + 32 * k;
            const float d = fabsf(px * X[k] + py * Y[k] + pz * Z[k]);
            if (d > CTH || i == cand0) c[k] = NEG_INF;
        }
    }

    // ---- Pass 1: argmax over survivors.
    bv = NEG_INF; bi = 0;
    #pragma unroll
    for (int k = 0; k < ITEMS_PER_LANE; ++k)
        if (c[k] > bv) { bv = c[k]; bi = lane + 32 * k; }
    wave_argmax(bv, bi);

    if (bv > NEG_INF) {           // a second VP exists
        cand1 = bi;
        const float px = wpos[cand1 * 3 + 0];
        const float py = wpos[cand1 * 3 + 1];
        const float pz = wpos[cand1 * 3 + 2];
        #pragma unroll
        for (int k = 0; k < ITEMS_PER_LANE; ++k) {
            const int   i = lane + 32 * k;
            const float d = fabsf(px * X[k] + py * Y[k] + pz * Z[k]);
            if (d > CTH || i == cand1) c[k] = NEG_INF;
        }

        // ---- Pass 2: argmax over items far from both cand0 and cand1.
        bv = NEG_INF; bi = 0;
        #pragma unroll
        for (int k = 0; k < ITEMS_PER_LANE; ++k)
            if (c[k] > bv) { bv = c[k]; bi = lane + 32 * k; }
        wave_argmax(bv, bi);
        if (bv > NEG_INF) cand2 = bi;
    }
    // If pass 1 found nothing, pass 2 cannot either (its filter is stricter):
    // slots 1 and 2 stay at index 0, matching the reference's zero-init cand.

    // ---- Emit pos[cand] : 9 floats per image, lanes 0..8.
    if (lane < NUM_VP * 3) {
        const int s  = lane / 3;          // which selected VP
        const int j  = lane - s * 3;      // which component
        const int ci = (s == 0) ? cand0 : ((s == 1) ? cand1 : cand2);
        out[(size_t)b * (NUM_VP * 3) + lane] = wpos[ci * 3 + j];
    }
}

extern "C" void kernel_launch(void* const* d_in, const int* in_sizes, int n_in,
                              void* d_out, int out_size, void* d_ws, size_t ws_size,
                              hipStream_t stream) {
    const float* logits = (const float*)d_in[0];   // [B, N, 1] f32
    const float* pos    = (const float*)d_in[1];   // [B, N, 3] f32
    // d_in[2] (vps_unique) only fixes num=3; it does not affect the output.
    float* out = (float*)d_out;                    // [B, 3, 3] f32

    const int B      = in_sizes[0] / N_Q;
    const int blocks = (B + WAVES_PER_BLK - 1) / WAVES_PER_BLK;

    hipLaunchKernelGGL(vp_greedy_select_kernel,
                       dim3(blocks), dim3(WAVES_PER_BLK * 32), 0, stream,
                       logits, pos, out, B);
}